// Denoiser_38671885533631
// MI455X (gfx1250) — compile-verified
//
#include <hip/hip_runtime.h>
#include <math.h>

#define NN   4096
#define FS   64
#define FV   32
#define KK   32
#define NBIN 32
#define NLAY 3
#define OUTW 163      // FS + FV*3 + 3
#define RCUT 32.0f

typedef __attribute__((ext_vector_type(2))) float v2f;
typedef __attribute__((ext_vector_type(8))) float v8f;

__device__ __forceinline__ v8f wmma4(v2f a, v2f b, v8f c) {
  // D = A(16x4,f32) * B(4x16,f32) + C(16x16,f32)
  return __builtin_amdgcn_wmma_f32_16x16x4_f32(false, a, false, b, (short)0, c,
                                               false, false);
}

// --- CDNA5 async global->LDS copies (ASYNCcnt path) ----------------------
__device__ __forceinline__ unsigned lds_of(const void* p) {
  // low 32 bits of a generic pointer into LDS == LDS byte address
  return (unsigned)(uintptr_t)p;
}
__device__ __forceinline__ void async_b128(unsigned lds, const float* g) {
  asm volatile("global_load_async_to_lds_b128 %0, %1, off"
               :: "v"(lds), "v"(g) : "memory");
}
__device__ __forceinline__ void async_b32(unsigned lds, const float* g) {
  asm volatile("global_load_async_to_lds_b32 %0, %1, off"
               :: "v"(lds), "v"(g) : "memory");
}
__device__ __forceinline__ void wait_async() {
  asm volatile("s_wait_asynccnt 0x0" ::: "memory");
}

// Stage W[Kd x Nd] (row-major, contiguous) into LDS in (k,k+1)-pair-major
// layout: shP[(p*Nd + c)*2 + {0,1}] = {W[2p][c], W[2p+1][c]} so a B fragment
// {W[ka][col], W[ka+1][col]} is one contiguous 8B-aligned ds_load_b64.
template <int KD, int ND>
__device__ __forceinline__ void stage_B_packed(float* shP, const float* Wm,
                                               int tid, int nthr) {
  for (int i = tid; i < (KD / 2) * (ND / 2); i += nthr) {
    const int p = i / (ND / 2);
    const int c = (i - p * (ND / 2)) * 2;
    v2f r0 = *(const v2f*)(Wm + (2 * p) * ND + c);      // {W[2p][c], W[2p][c+1]}
    v2f r1 = *(const v2f*)(Wm + (2 * p + 1) * ND + c);  // {W[2p+1][c], ...}
    v2f s0, s1;
    s0.x = r0.x; s0.y = r1.x;
    s1.x = r0.y; s1.y = r1.y;
    *(v2f*)(shP + (p * ND + c) * 2) = s0;
    *(v2f*)(shP + (p * ND + c + 1) * 2) = s1;
  }
}

// ---------------------------------------------------------------------------
// kNN graph: one block (256 thr) per node. Full d2 row in LDS, 32 rounds of
// block-wide argmin. Emits idx, unit vectors, and RBF features.
// ---------------------------------------------------------------------------
__global__ void knn_kernel(const float* __restrict__ coord,
                           int* __restrict__ idx_out,
                           float* __restrict__ unit_out,
                           float* __restrict__ rbf_out) {
  __shared__ float d2[NN];
  __shared__ float rmin[256];
  __shared__ int   rarg[256];
  __shared__ int   sel[KK];
  const int i = blockIdx.x;
  const int tid = threadIdx.x;
  const float cx = coord[i * 3 + 0];
  const float cy = coord[i * 3 + 1];
  const float cz = coord[i * 3 + 2];
  for (int j = tid; j < NN; j += 256) {
    float dx = coord[j * 3 + 0] - cx;
    float dy = coord[j * 3 + 1] - cy;
    float dz = coord[j * 3 + 2] - cz;
    float dd = dx * dx + dy * dy + dz * dz;
    d2[j] = (j == i) ? 1e30f : dd;
  }
  __syncthreads();
  for (int k = 0; k < KK; ++k) {
    float best = 1e30f;
    int barg = NN;
    for (int j = tid; j < NN; j += 256) {
      float v = d2[j];
      if (v < best || (v == best && j < barg)) { best = v; barg = j; }
    }
    rmin[tid] = best;
    rarg[tid] = barg;
    __syncthreads();
    for (int off = 128; off > 0; off >>= 1) {
      if (tid < off) {
        float ov = rmin[tid + off];
        int oa = rarg[tid + off];
        if (ov < rmin[tid] || (ov == rmin[tid] && oa < rarg[tid])) {
          rmin[tid] = ov;
          rarg[tid] = oa;
        }
      }
      __syncthreads();
    }
    if (tid == 0) {
      sel[k] = rarg[0];
      d2[rarg[0]] = 1e30f;
    }
    __syncthreads();
  }
  if (tid < KK) {
    int j = sel[tid];
    idx_out[i * KK + tid] = j;
    float rx = coord[j * 3 + 0] - cx;
    float ry = coord[j * 3 + 1] - cy;
    float rz = coord[j * 3 + 2] - cz;
    float d = sqrtf(rx * rx + ry * ry + rz * rz + 1e-12f);
    float inv = 1.0f / (d + 1e-9f);
    unit_out[(i * KK + tid) * 3 + 0] = rx * inv;
    unit_out[(i * KK + tid) * 3 + 1] = ry * inv;
    unit_out[(i * KK + tid) * 3 + 2] = rz * inv;
    const float width = RCUT / (float)NBIN;  // = 1.0
    for (int b = 0; b < NBIN; ++b) {
      float c = (float)b * (RCUT / (float)(NBIN - 1));
      float x = (d - c) / width;
      rbf_out[((size_t)i * KK + tid) * NBIN + b] = __expf(-x * x);
    }
  }
}

// ---------------------------------------------------------------------------
// Scalar-channel GEMM: Y[4096,64](yr-strided rows) = X[4096,64] * W[64,64].
// B pre-swizzled pair-major in LDS; one wave = 16-row strip x 4 col tiles.
// Inner loop: 1 global b64 (A) + 4 ds_load_b64 (B) + 4 v_wmma per k-step.
// ---------------------------------------------------------------------------
__global__ void gemm_s_kernel(const float* __restrict__ X,
                              const float* __restrict__ Wm,
                              float* __restrict__ Y, int yr, int accumulate) {
  __shared__ __align__(16) float shB[FS * FS];  // 16 KB, pair-major
  const int tid = threadIdx.x;
  const int lane = tid & 31;
  const int wave = tid >> 5;
  const int half = lane >> 4;
  const int l15 = lane & 15;

  stage_B_packed<FS, FS>(shB, Wm, tid, 256);

  const int mbase = (blockIdx.x * 8 + wave) * 16;
  v8f acc[4];
  if (accumulate) {
#pragma unroll
    for (int t = 0; t < 4; ++t)
#pragma unroll
      for (int j = 0; j < 8; ++j)
        acc[t][j] = Y[(size_t)(mbase + j + 8 * half) * yr + t * 16 + l15];
  } else {
#pragma unroll
    for (int t = 0; t < 4; ++t)
#pragma unroll
      for (int j = 0; j < 8; ++j) acc[t][j] = 0.0f;
  }
  __syncthreads();

  const float* Xr = X + (size_t)(mbase + l15) * FS;
#pragma unroll
  for (int k0 = 0; k0 < FS; k0 += 4) {
    const int ka = k0 + 2 * half;
    const int p = ka >> 1;
    v2f a = *(const v2f*)(Xr + ka);  // 8B-aligned contiguous pair
#pragma unroll
    for (int t = 0; t < 4; ++t) {
      v2f bm = *(const v2f*)(shB + (p * FS + t * 16 + l15) * 2);
      acc[t] = wmma4(a, bm, acc[t]);
    }
  }
#pragma unroll
  for (int t = 0; t < 4; ++t)
#pragma unroll
    for (int j = 0; j < 8; ++j)
      Y[(size_t)(mbase + j + 8 * half) * yr + t * 16 + l15] = acc[t][j];
}

// ---------------------------------------------------------------------------
// Vector-channel GEMM, batched over d=0..2 (grid.z):
// Y[n, g, d] = sum_f X[n, f, d] * W[f, g];  X strides (96,3), Y (yr,3).
// ---------------------------------------------------------------------------
__global__ void gemm_v_kernel(const float* __restrict__ X,
                              const float* __restrict__ Wm,
                              float* __restrict__ Y, int yr, int accumulate) {
  __shared__ __align__(16) float shB[FV * FV];  // 4 KB, pair-major
  const int tid = threadIdx.x;
  const int lane = tid & 31;
  const int wave = tid >> 5;
  const int half = lane >> 4;
  const int l15 = lane & 15;
  const int d = blockIdx.z;

  stage_B_packed<FV, FV>(shB, Wm, tid, 256);

  const int mbase = (blockIdx.x * 8 + wave) * 16;
  v8f acc[2];
  if (accumulate) {
#pragma unroll
    for (int t = 0; t < 2; ++t)
#pragma unroll
      for (int j = 0; j < 8; ++j)
        acc[t][j] = Y[(size_t)(mbase + j + 8 * half) * yr + (t * 16 + l15) * 3 + d];
  } else {
#pragma unroll
    for (int t = 0; t < 2; ++t)
#pragma unroll
      for (int j = 0; j < 8; ++j) acc[t][j] = 0.0f;
  }
  __syncthreads();

  const float* Xr = X + (size_t)(mbase + l15) * (FV * 3) + d;
#pragma unroll
  for (int k0 = 0; k0 < FV; k0 += 4) {
    const int ka = k0 + 2 * half;
    const int p = ka >> 1;
    v2f a;
    a.x = Xr[ka * 3];
    a.y = Xr[ka * 3 + 3];
#pragma unroll
    for (int t = 0; t < 2; ++t) {
      v2f bm = *(const v2f*)(shB + (p * FV + t * 16 + l15) * 2);
      acc[t] = wmma4(a, bm, acc[t]);
    }
  }
#pragma unroll
  for (int t = 0; t < 2; ++t)
#pragma unroll
    for (int j = 0; j < 8; ++j)
      Y[(size_t)(mbase + j + 8 * half) * yr + (t * 16 + l15) * 3 + d] = acc[t][j];
}

// ---------------------------------------------------------------------------
// s[n,:] += time_emb[t[n],:]
// ---------------------------------------------------------------------------
__global__ void add_emb_kernel(float* __restrict__ s, const int* __restrict__ t,
                               const float* __restrict__ emb) {
  int i = blockIdx.x * blockDim.x + threadIdx.x;
  if (i >= NN * FS) return;
  int n = i >> 6;
  int f = i & 63;
  s[i] += emb[t[n] * FS + f];
}

// ---------------------------------------------------------------------------
// Equivariant LayerNorm, in place. One wave per node (wave32).
// ---------------------------------------------------------------------------
__global__ void eln_kernel(float* __restrict__ s, float* __restrict__ v) {
  const int node = blockIdx.x * (blockDim.x >> 5) + (threadIdx.x >> 5);
  const int lane = threadIdx.x & 31;
  float a0 = s[node * FS + lane];
  float a1 = s[node * FS + 32 + lane];
  float sum = a0 + a1;
#pragma unroll
  for (int o = 16; o >= 1; o >>= 1) sum += __shfl_xor(sum, o, 32);
  float mean = sum * (1.0f / (float)FS);
  float d0 = a0 - mean, d1 = a1 - mean;
  float ss = d0 * d0 + d1 * d1;
#pragma unroll
  for (int o = 16; o >= 1; o >>= 1) ss += __shfl_xor(ss, o, 32);
  float inv = rsqrtf(ss * (1.0f / (float)FS) + 1e-6f);
  s[node * FS + lane] = d0 * inv;
  s[node * FS + 32 + lane] = d1 * inv;
  float vx = v[node * FV * 3 + lane * 3 + 0];
  float vy = v[node * FV * 3 + lane * 3 + 1];
  float vz = v[node * FV * 3 + lane * 3 + 2];
  float q = vx * vx + vy * vy + vz * vz;
#pragma unroll
  for (int o = 16; o >= 1; o >>= 1) q += __shfl_xor(q, o, 32);
  float vn = sqrtf(q * (1.0f / (float)FV) + 1e-6f);
  float ivn = 1.0f / vn;
  v[node * FV * 3 + lane * 3 + 0] = vx * ivn;
  v[node * FV * 3 + lane * 3 + 1] = vy * ivn;
  v[node * FV * 3 + lane * 3 + 2] = vz * ivn;
}

// ---------------------------------------------------------------------------
// Fused kNN spatial conv: one block (8 waves) per node. rbf/sn/vn/unit are
// staged with CDNA5 async global->LDS (gathers included); radial weights are
// staged pair-major for mov-free ds_load_b64 B fragments; ws/wv/wsv computed
// with WMMA; elementwise * gathered features + mean-over-K via ds_add_f32.
// ---------------------------------------------------------------------------
__global__ void conv_kernel(const float* __restrict__ s1,    // [N,FS]
                            const float* __restrict__ v1,    // [N,FV,3]
                            const int* __restrict__ idx,     // [N,K]
                            const float* __restrict__ unit,  // [N,K,3]
                            const float* __restrict__ rbf,   // [N,K,NBIN]
                            const float* __restrict__ Wrs,   // [NBIN,FS]
                            const float* __restrict__ Wrv,   // [NBIN,FV]
                            const float* __restrict__ Wrsv,  // [NBIN,FV]
                            float* __restrict__ ms,          // [N,FS]
                            float* __restrict__ mv) {        // [N,FV,3]
  __shared__ __align__(16) float sh_rbf[KK * NBIN];   // 4 KB
  __shared__ __align__(16) float sh_w[4096];          // packed Wrs|Wrv|Wrsv
  __shared__ __align__(16) float sh_sn[KK * FS];      // 8 KB
  __shared__ __align__(16) float sh_vn[KK * FV * 3];  // 12 KB
  __shared__ __align__(16) float sh_unit[KK * 3];
  __shared__ float sh_ms[FS];
  __shared__ float sh_mv[FV * 3];
  __shared__ int sh_idx[KK];

  const int n = blockIdx.x;
  const int tid = threadIdx.x;
  const int lane = tid & 31;
  const int wave = tid >> 5;

  if (tid < KK) sh_idx[tid] = idx[n * KK + tid];
  if (tid < FS) sh_ms[tid] = 0.0f;
  if (tid < FV * 3) sh_mv[tid] = 0.0f;
  __syncthreads();

  // --- async staging: rbf + gathered sn/vn + unit ---
  async_b128(lds_of(&sh_rbf[tid * 4]), rbf + (size_t)n * KK * NBIN + tid * 4);
  for (int i = tid; i < 512; i += 256) {   // sn: 32 rows x 64 f
    int k = i >> 4, c = (i & 15) * 4;
    async_b128(lds_of(&sh_sn[k * FS + c]), s1 + (size_t)sh_idx[k] * FS + c);
  }
  for (int i = tid; i < 768; i += 256) {   // vn: 32 rows x 96 f
    int k = i / 24, c = (i - k * 24) * 4;
    async_b128(lds_of(&sh_vn[k * FV * 3 + c]), v1 + (size_t)sh_idx[k] * FV * 3 + c);
  }
  if (tid < KK * 3)
    async_b32(lds_of(&sh_unit[tid]), unit + (size_t)n * KK * 3 + tid);
  // --- pair-major radial weights (direct swizzled staging) ---
  stage_B_packed<NBIN, FS>(sh_w, Wrs, tid, 256);         // [0, 2048)
  stage_B_packed<NBIN, FV>(sh_w + 2048, Wrv, tid, 256);  // [2048, 3072)
  stage_B_packed<NBIN, FV>(sh_w + 3072, Wrsv, tid, 256); // [3072, 4096)
  wait_async();
  __syncthreads();

  const int half = lane >> 4;
  const int l15 = lane & 15;
  // 16 tiles: 0..7 -> ws (2x4), 8..11 -> wv (2x2), 12..15 -> wsv (2x2)
#pragma unroll
  for (int p = 0; p < 2; ++p) {
    const int t = wave * 2 + p;
    int tm, tn, kind, wbase, ldw;
    if (t < 8) { kind = 0; tm = t >> 2; tn = t & 3; wbase = 0; ldw = FS; }
    else if (t < 12) { kind = 1; int tt = t - 8; tm = tt >> 1; tn = tt & 1; wbase = 2048; ldw = FV; }
    else { kind = 2; int tt = t - 12; tm = tt >> 1; tn = tt & 1; wbase = 3072; ldw = FV; }

    const int arow = tm * 16 + l15;  // neighbor index k
    const int bcol = tn * 16 + l15;  // feature index f
    v8f acc;
#pragma unroll
    for (int j = 0; j < 8; ++j) acc[j] = 0.0f;
#pragma unroll
    for (int kb = 0; kb < NBIN; kb += 4) {
      const int ka = kb + 2 * half;
      v2f a = *(const v2f*)(sh_rbf + arow * NBIN + ka);               // ds_load_b64
      v2f bm = *(const v2f*)(sh_w + wbase + ((ka >> 1) * ldw + bcol) * 2);
      acc = wmma4(a, bm, acc);
    }
    if (kind == 0) {
      float part = 0.0f;
#pragma unroll
      for (int j = 0; j < 8; ++j) {
        int k = tm * 16 + j + 8 * half;
        part += acc[j] * sh_sn[k * FS + bcol];
      }
      atomicAdd(&sh_ms[bcol], part);
    } else if (kind == 1) {
      float p0 = 0.0f, p1 = 0.0f, p2 = 0.0f;
#pragma unroll
      for (int j = 0; j < 8; ++j) {
        int k = tm * 16 + j + 8 * half;
        p0 += acc[j] * sh_vn[k * FV * 3 + bcol * 3 + 0];
        p1 += acc[j] * sh_vn[k * FV * 3 + bcol * 3 + 1];
        p2 += acc[j] * sh_vn[k * FV * 3 + bcol * 3 + 2];
      }
      atomicAdd(&sh_mv[bcol * 3 + 0], p0);
      atomicAdd(&sh_mv[bcol * 3 + 1], p1);
      atomicAdd(&sh_mv[bcol * 3 + 2], p2);
    } else {
      float p0 = 0.0f, p1 = 0.0f, p2 = 0.0f;
#pragma unroll
      for (int j = 0; j < 8; ++j) {
        int k = tm * 16 + j + 8 * half;
        p0 += acc[j] * sh_unit[k * 3 + 0];
        p1 += acc[j] * sh_unit[k * 3 + 1];
        p2 += acc[j] * sh_unit[k * 3 + 2];
      }
      atomicAdd(&sh_mv[bcol * 3 + 0], p0);
      atomicAdd(&sh_mv[bcol * 3 + 1], p1);
      atomicAdd(&sh_mv[bcol * 3 + 2], p2);
    }
  }
  __syncthreads();
  const float invK = 1.0f / (float)KK;
  if (tid < FS) ms[(size_t)n * FS + tid] = sh_ms[tid] * invK;
  if (tid < FV * 3) mv[(size_t)n * FV * 3 + tid] = sh_mv[tid] * invK;
}

// ---------------------------------------------------------------------------
// coord_out[n,d] = sum_f hv[n,f,d] * Wcoord[f]
// ---------------------------------------------------------------------------
__global__ void coord_out_kernel(const float* __restrict__ hv,
                                 const float* __restrict__ Wc,
                                 float* __restrict__ out) {
  int i = blockIdx.x * blockDim.x + threadIdx.x;
  if (i >= NN * 3) return;
  int n = i / 3, d = i - n * 3;
  float acc = 0.0f;
#pragma unroll
  for (int f = 0; f < FV; ++f) acc += hv[(size_t)n * FV * 3 + f * 3 + d] * Wc[f];
  out[(size_t)n * OUTW + FS + FV * 3 + d] = acc;
}

// ---------------------------------------------------------------------------
extern "C" void kernel_launch(void* const* d_in, const int* in_sizes, int n_in,
                              void* d_out, int out_size, void* d_ws, size_t ws_size,
                              hipStream_t stream) {
  const float* coord   = (const float*)d_in[0];
  const float* s_in    = (const float*)d_in[1];
  const float* v_in    = (const float*)d_in[2];
  const int*   t_in    = (const int*)d_in[3];
  const float* temb    = (const float*)d_in[4];
  const float* W_si0_s = (const float*)d_in[5];
  const float* W_si0_v = (const float*)d_in[6];
  const float* W_si1_s = (const float*)d_in[7];
  const float* W_si1_v = (const float*)d_in[8];
  const float* Wsi_s   = (const float*)d_in[9];
  const float* Wsi_v   = (const float*)d_in[10];
  const float* Wrad_s  = (const float*)d_in[11];
  const float* Wrad_v  = (const float*)d_in[12];
  const float* Wrad_sv = (const float*)d_in[13];
  const float* Wres_s  = (const float*)d_in[14];
  const float* Wres_v  = (const float*)d_in[15];
  const float* Wfin_s  = (const float*)d_in[16];
  const float* Wfin_v  = (const float*)d_in[17];
  const float* Wout_s  = (const float*)d_in[18];
  const float* Wout_v  = (const float*)d_in[19];
  const float* Wcoord  = (const float*)d_in[20];
  float* out = (float*)d_out;

  // workspace layout (4-byte units)
  float* WSP = (float*)d_ws;
  size_t o = 0;
  int*   w_idx  = (int*)(WSP + o); o += (size_t)NN * KK;
  float* w_unit = WSP + o;         o += (size_t)NN * KK * 3;
  float* w_rbf  = WSP + o;         o += (size_t)NN * KK * NBIN;
  float* st_s   = WSP + o;         o += (size_t)(NLAY + 1) * NN * FS;
  float* st_v   = WSP + o;         o += (size_t)(NLAY + 1) * NN * FV * 3;
  float* stmp   = WSP + o;         o += (size_t)NN * FS;
  float* vtmp   = WSP + o;         o += (size_t)NN * FV * 3;
  float* s1     = WSP + o;         o += (size_t)NN * FS;
  float* v1b    = WSP + o;         o += (size_t)NN * FV * 3;
  float* msb    = WSP + o;         o += (size_t)NN * FS;
  float* mvb    = WSP + o;         o += (size_t)NN * FV * 3;
  float* hs     = WSP + o;         o += (size_t)NN * FS;
  float* hv     = WSP + o;         o += (size_t)NN * FV * 3;

  const dim3 gs(NN / 16 / 8, 1, 1);   // scalar GEMM grid (32 blocks)
  const dim3 gv(NN / 16 / 8, 1, 3);   // vector GEMM grid (x3 components)

  auto gemm_s = [&](const float* X, const float* Wm, float* Y, int yr, int acc) {
    gemm_s_kernel<<<gs, 256, 0, stream>>>(X, Wm, Y, yr, acc);
  };
  auto gemm_v = [&](const float* X, const float* Wm, float* Y, int yr, int acc) {
    gemm_v_kernel<<<gv, 256, 0, stream>>>(X, Wm, Y, yr, acc);
  };

  // --- kNN graph + RBF ---
  knn_kernel<<<NN, 256, 0, stream>>>(coord, w_idx, w_unit, w_rbf);

  // --- initial self-interaction + time embedding ---
  gemm_s(s_in, W_si0_s, stmp, FS, 0);
  add_emb_kernel<<<(NN * FS + 255) / 256, 256, 0, stream>>>(stmp, t_in, temb);
  gemm_s(stmp, W_si1_s, st_s, FS, 0);
  gemm_v(v_in, W_si0_v, vtmp, FV * 3, 0);
  gemm_v(vtmp, W_si1_v, st_v, FV * 3, 0);
  eln_kernel<<<NN / 8, 256, 0, stream>>>(st_s, st_v);

  float* s_cur = st_s;
  float* v_cur = st_v;
  for (int l = 0; l < NLAY; ++l) {
    gemm_s(s_cur, Wsi_s + (size_t)l * FS * FS, s1, FS, 0);
    gemm_v(v_cur, Wsi_v + (size_t)l * FV * FV, v1b, FV * 3, 0);
    eln_kernel<<<NN / 8, 256, 0, stream>>>(s1, v1b);
    conv_kernel<<<NN, 256, 0, stream>>>(s1, v1b, w_idx, w_unit, w_rbf,
                                        Wrad_s + (size_t)l * NBIN * FS,
                                        Wrad_v + (size_t)l * NBIN * FV,
                                        Wrad_sv + (size_t)l * NBIN * FV,
                                        msb, mvb);
    float* s_nxt = st_s + (size_t)(l + 1) * NN * FS;
    float* v_nxt = st_v + (size_t)(l + 1) * NN * FV * 3;
    const float* Wrs_l = Wres_s + (size_t)l * 2 * FS * FS;
    const float* Wrv_l = Wres_v + (size_t)l * 2 * FV * FV;
    gemm_s(s1, Wrs_l, s_nxt, FS, 0);
    gemm_s(msb, Wrs_l + (size_t)FS * FS, s_nxt, FS, 1);
    gemm_v(v1b, Wrv_l, v_nxt, FV * 3, 0);
    gemm_v(mvb, Wrv_l + (size_t)FV * FV, v_nxt, FV * 3, 1);
    eln_kernel<<<NN / 8, 256, 0, stream>>>(s_nxt, v_nxt);
    s_cur = s_nxt;
    v_cur = v_nxt;
  }

  // --- final linear over concatenated states (accumulating slice GEMMs) ---
  for (int i = 0; i < NLAY + 1; ++i) {
    gemm_s(st_s + (size_t)i * NN * FS, Wfin_s + (size_t)i * FS * FS, hs, FS, i > 0);
    gemm_v(st_v + (size_t)i * NN * FV * 3, Wfin_v + (size_t)i * FV * FV, hv, FV * 3, i > 0);
  }

  // --- outputs straight into d_out [N, 163] ---
  gemm_s(hs, Wout_s, out, OUTW, 0);
  gemm_v(hv, Wout_v, out + FS, OUTW, 0);
  coord_out_kernel<<<(NN * 3 + 255) / 256, 256, 0, stream>>>(hv, Wcoord, out);
}